// GeometricEigenmodes_41583873360604
// MI455X (gfx1250) — compile-verified
//
#include <hip/hip_runtime.h>
#include <math.h>

// ---------------------------------------------------------------------------
// GeometricEigenmodes on MI455X (gfx1250, wave32, WMMA + TDM)
//
//  1. zero L (64MB dense Laplacian -> resident in 192MB L2), init X (4096x112)
//  2. scatter cotangent weights into L (atomics), add 1e-6*I
//  3. Gershgorin shift s = max_r sum_c |L[r][c]|  (device-side atomicMax)
//  4. subspace iteration:  Y = s*X - L*X  (v_wmma_f32_16x16x4_f32 GEMM; L
//     fragments streamed from L2, X panel DMA'd into double-buffered LDS by
//     the Tensor Data Mover, synced with s_wait_tensorcnt), Cholesky-QR
//     re-orthonormalization (Gram = X^T X via WMMA)
//  5. Rayleigh-Ritz: W = L*X, Ar = X^T W (WMMA), Jacobi eig of 112x112,
//     sort ascending, X <- X*Q (WMMA), column-normalize first 100 -> d_out
// ---------------------------------------------------------------------------

typedef __attribute__((ext_vector_type(2))) float    v2f;
typedef __attribute__((ext_vector_type(8))) float    v8f;
typedef __attribute__((ext_vector_type(4))) unsigned v4u;
typedef __attribute__((ext_vector_type(4))) int      v4i;
typedef __attribute__((ext_vector_type(8))) int      v8i;

#define NV 4096      // vertices
#define NF 8192      // faces
#define KM 112       // padded mode count (7 WMMA tiles of 16)
#define NMODES 100
#define EPSF 1e-8f
#define ITERS 32

#if defined(__gfx1250__) && __has_builtin(__builtin_amdgcn_tensor_load_to_lds)
#define USE_TDM 1
#else
#define USE_TDM 0
#endif

// ---------------------------------------------------------------- utilities
__global__ void k_zero(float* p, long n) {
  long i = (long)blockIdx.x * blockDim.x + threadIdx.x;
  long stride = (long)gridDim.x * blockDim.x;
  for (; i < n; i += stride) p[i] = 0.0f;
}

__global__ void k_initX(float* X) {
  int i = blockIdx.x * blockDim.x + threadIdx.x;
  if (i >= NV * KM) return;
  unsigned h = (unsigned)i * 2654435761u;
  h ^= h >> 16; h *= 2246822519u; h ^= h >> 13;
  X[i] = ((h & 0xFFFFFF) * (1.0f / 16777216.0f)) - 0.5f;
}

// -------------------------------------------------- cotangent scatter into L
__device__ __forceinline__ float cot3(const float* p, const float* q, const float* r) {
  float ux = q[0]-p[0], uy = q[1]-p[1], uz = q[2]-p[2];
  float vx = r[0]-p[0], vy = r[1]-p[1], vz = r[2]-p[2];
  float dot = ux*vx + uy*vy + uz*vz;
  float nu = sqrtf(ux*ux + uy*uy + uz*uz);
  float nv = sqrtf(vx*vx + vy*vy + vz*vz);
  float c = dot / (nu * nv + EPSF);
  float s2 = 1.0f - c * c;
  return c / (sqrtf(fmaxf(s2, 0.0f)) + EPSF);
}

__global__ void k_scatter(const float* __restrict__ V, const int* __restrict__ F,
                          float* __restrict__ Lm) {
  int f = blockIdx.x * blockDim.x + threadIdx.x;
  if (f >= NF) return;
  int a = F[3*f+0], b = F[3*f+1], c = F[3*f+2];
  float Va[3] = {V[3*a], V[3*a+1], V[3*a+2]};
  float Vb[3] = {V[3*b], V[3*b+1], V[3*b+2]};
  float Vc[3] = {V[3*c], V[3*c+1], V[3*c+2]};
  float cab = cot3(Va, Vb, Vc);
  float cba = cot3(Vb, Va, Vc);
  float ccb = cot3(Vc, Va, Vb);
  atomicAdd(&Lm[(size_t)a*NV + b], -cab);
  atomicAdd(&Lm[(size_t)b*NV + a], -cab);
  atomicAdd(&Lm[(size_t)a*NV + c], -cba);
  atomicAdd(&Lm[(size_t)c*NV + a], -cba);
  atomicAdd(&Lm[(size_t)b*NV + c], -ccb);
  atomicAdd(&Lm[(size_t)c*NV + b], -ccb);
  atomicAdd(&Lm[(size_t)a*NV + a], cab + cba);
  atomicAdd(&Lm[(size_t)b*NV + b], cab + ccb);
  atomicAdd(&Lm[(size_t)c*NV + c], cba + ccb);
}

// diag += 1e-6, then Gershgorin row-abs-sum -> atomicMax(shift) (float bits)
__global__ __launch_bounds__(256) void k_rowshift(float* __restrict__ Lm, float* shift) {
  __shared__ float red[256];
  int r = blockIdx.x;
  if (threadIdx.x == 0) Lm[(size_t)r*NV + r] += 1e-6f;
  __syncthreads();
  float s = 0.0f;
  for (int c = threadIdx.x; c < NV; c += 256) s += fabsf(Lm[(size_t)r*NV + c]);
  red[threadIdx.x] = s;
  __syncthreads();
  for (int o = 128; o > 0; o >>= 1) {
    if (threadIdx.x < o) red[threadIdx.x] += red[threadIdx.x + o];
    __syncthreads();
  }
  if (threadIdx.x == 0) atomicMax((int*)shift, __float_as_int(red[0]));
}

#if USE_TDM
// Issue a TDM 2D-tile load: 64 rows x 16 cols (f32) of X, row stride KM,
// packed densely into LDS at ldsaddr. D# packed per CDNA5 ISA 8.3/8.4.
// clang-23 / therock-10.0 builtin arity: (v4u, v8i, v4i, v4i, v8i, i32 cpol).
__device__ __forceinline__ void tdm_load_panel(const float* gptr, unsigned ldsaddr) {
  unsigned long long ga = (unsigned long long)(size_t)gptr;
  v4u g0;
  g0.x = 1u;                                   // count=1 valid descriptor
  g0.y = ldsaddr;                              // lds_addr (bytes)
  g0.z = (unsigned)(ga & 0xFFFFFFFFu);         // global_addr[31:0]
  g0.w = (unsigned)((ga >> 32) & 0x01FFFFFFu)  // global_addr[56:32]
       | (2u << 30);                           // type = 2 ("image")
  v8i g1;
  g1[0] = 0x00020000;                          // data_size=2 (4B), no mask
  g1[1] = (int)(112u << 16);                   // tensor_dim0 = 112 (KM)
  g1[2] = (int)(4096u << 16);                  // tensor_dim1 = 4096 (NV)
  g1[3] = (int)(16u << 16);                    // tile_dim0 = 16
  g1[4] = 64;                                  // tile_dim1 = 64, tile_dim2 = 0
  g1[5] = 112;                                 // tensor_dim0_stride = KM
  g1[6] = 0;
  g1[7] = 0;                                   // tensor_dim1_stride unused (2D)
  v4i g2 = {0, 0, 0, 0};                       // group2: unused (2D tensor)
  v4i g3 = {0, 0, 0, 0};                       // group3: unused (2D tensor)
  v8i g4 = {0, 0, 0, 0, 0, 0, 0, 0};           // extra group (zero-filled)
  __builtin_amdgcn_tensor_load_to_lds(g0, g1, g2, g3, g4, 0);
}
#endif

// ---------------------------------------------------------------------------
// Out(4096xKM) = alpha*Xin + beta*(L @ Xin),  alpha = *shiftPtr (or 0)
// grid(32,7), 256 threads = 8 waves; wave w owns a 16x16 output tile.
// X panel double-buffered in LDS, filled by the Tensor Data Mover (TDM);
// L rows streamed from L2 with explicit prefetch of the next panel.
// ---------------------------------------------------------------------------
__global__ __launch_bounds__(256) void k_gemmLX(const float* __restrict__ Lm,
                                                const float* __restrict__ Xin,
                                                float* __restrict__ Out,
                                                const float* __restrict__ shiftPtr,
                                                float beta) {
  __shared__ float tX[2][64][16];
  const int lane = threadIdx.x & 31;
  const int wave = threadIdx.x >> 5;
  const int m    = lane & 15;
  const int half = lane >> 4;
  const int row0 = blockIdx.x * 128 + wave * 16;
  const int col0 = blockIdx.y * 16;

#if USE_TDM
  if (wave == 0) {  // prologue: DMA stage 0 into buffer 0
    tdm_load_panel(Xin + (size_t)0 * KM + col0, (unsigned)(size_t)&tX[0][0][0]);
  }
#endif

  v8f acc = {};
  for (int s = 0; s < NV / 64; ++s) {
    const int k0 = s * 64;
    const int buf = s & 1;
#if USE_TDM
    if (wave == 0) {
      if (k0 + 64 < NV) {  // DMA next stage into the other buffer
        tdm_load_panel(Xin + (size_t)(k0 + 64) * KM + col0,
                       (unsigned)(size_t)&tX[buf ^ 1][0][0]);
        __builtin_amdgcn_s_wait_tensorcnt(1);   // current stage resident
      } else {
        __builtin_amdgcn_s_wait_tensorcnt(0);
      }
    }
    __syncthreads();
#else
    for (int t = threadIdx.x; t < 64 * 16; t += 256) {
      int kk = t >> 4, c = t & 15;
      tX[buf][kk][c] = Xin[(size_t)(k0 + kk) * KM + col0 + c];
    }
    __syncthreads();
#endif
    if (k0 + 64 < NV) {  // prefetch next L panel for this wave's rows
      __builtin_prefetch(Lm + (size_t)(row0 + m) * NV + (k0 + 64) + half * 32, 0, 1);
    }
#pragma unroll
    for (int kk = 0; kk < 64; kk += 4) {
      const float* ap = Lm + (size_t)(row0 + m) * NV + k0 + kk + 2 * half;
      v2f a = *reinterpret_cast<const v2f*>(ap);   // 8B aligned
      v2f b;
      b.x = tX[buf][kk + 2 * half + 0][m];
      b.y = tX[buf][kk + 2 * half + 1][m];
      acc = __builtin_amdgcn_wmma_f32_16x16x4_f32(false, a, false, b,
                                                  (short)0, acc, false, false);
    }
    __syncthreads();   // stage done; its buffer becomes free for DMA overwrite
  }

  const bool useShift = (shiftPtr != nullptr);
  const float alpha = useShift ? shiftPtr[0] : 0.0f;
#pragma unroll
  for (int v = 0; v < 8; ++v) {
    int r = row0 + v + 8 * half;   // D layout: VGPR v -> M = v + 8*half
    int c = col0 + m;              //           lane  -> N
    float xv = useShift ? Xin[(size_t)r * KM + c] : 0.0f;
    Out[(size_t)r * KM + c] = alpha * xv + beta * acc[v];
  }
}

// ---------------------------------------------------------------------------
// C(KMxKM) = A^T @ B,  A,B are 4096xKM.  grid(7,7), one wave per 16x16 tile.
// ---------------------------------------------------------------------------
__global__ __launch_bounds__(32) void k_atb(const float* __restrict__ A,
                                            const float* __restrict__ B,
                                            float* __restrict__ C) {
  const int lane = threadIdx.x & 31;
  const int m    = lane & 15;
  const int half = lane >> 4;
  const int rt = blockIdx.x * 16;
  const int ct = blockIdx.y * 16;
  v8f acc = {};
  for (int k = 0; k < NV; k += 4) {
    v2f a, b;
    a.x = A[(size_t)(k + 2 * half + 0) * KM + rt + m];   // A^T[rt+m][k..]
    a.y = A[(size_t)(k + 2 * half + 1) * KM + rt + m];
    b.x = B[(size_t)(k + 2 * half + 0) * KM + ct + m];
    b.y = B[(size_t)(k + 2 * half + 1) * KM + ct + m];
    acc = __builtin_amdgcn_wmma_f32_16x16x4_f32(false, a, false, b,
                                                (short)0, acc, false, false);
  }
#pragma unroll
  for (int v = 0; v < 8; ++v)
    C[(size_t)(rt + v + 8 * half) * KM + ct + m] = acc[v];
}

// ---------------------------------------------------------------------------
// In-place Cholesky of G (KMxKM): G -> upper R with R^T R = G. One block.
// ---------------------------------------------------------------------------
__global__ __launch_bounds__(128) void k_chol(float* __restrict__ G) {
  for (int j = 0; j < KM; ++j) {
    if (threadIdx.x == 0)
      G[j*KM + j] = sqrtf(fmaxf(G[j*KM + j], 1e-20f));
    __syncthreads();
    float dj = G[j*KM + j];
    for (int c = j + 1 + threadIdx.x; c < KM; c += blockDim.x)
      G[j*KM + c] /= dj;
    __syncthreads();
    for (int i = j + 1; i < KM; ++i) {
      float rji = G[j*KM + i];
      for (int c = i + threadIdx.x; c < KM; c += blockDim.x)
        G[i*KM + c] -= rji * G[j*KM + c];
    }
    __syncthreads();
  }
  for (int t = threadIdx.x; t < KM * KM; t += blockDim.x) {
    int i = t / KM, c = t % KM;
    if (c < i) G[t] = 0.0f;
  }
}

// Solve x R = y per row (forward substitution), in place on Y. R upper-tri.
__global__ __launch_bounds__(128) void k_trisolve(float* __restrict__ Y,
                                                  const float* __restrict__ R) {
  int r = blockIdx.x * blockDim.x + threadIdx.x;
  if (r >= NV) return;
  float x[KM];
  float* y = Y + (size_t)r * KM;
  for (int j = 0; j < KM; ++j) {
    float s = y[j];
    for (int i = 0; i < j; ++i) s -= x[i] * R[i*KM + j];
    x[j] = s / R[j*KM + j];
  }
  for (int j = 0; j < KM; ++j) y[j] = x[j];
}

// ---------------------------------------------------------------------------
// Cyclic Jacobi eigensolver for symmetric KMxKM A (destroyed). One block.
// Q gets eigenvectors (columns), w eigenvalues.
// ---------------------------------------------------------------------------
__global__ __launch_bounds__(128) void k_jacobi(float* __restrict__ A,
                                                float* __restrict__ Q,
                                                float* __restrict__ w) {
  const int t = threadIdx.x;
  __shared__ float sc, ss;
  __shared__ int sskip;
  for (int i = t; i < KM * KM; i += blockDim.x)
    Q[i] = (i / KM == i % KM) ? 1.0f : 0.0f;
  __syncthreads();
  for (int sweep = 0; sweep < 10; ++sweep) {
    for (int p = 0; p < KM - 1; ++p) {
      for (int q = p + 1; q < KM; ++q) {
        if (t == 0) {
          float apq = A[p*KM + q];
          if (fabsf(apq) < 1e-12f) {
            sskip = 1;
          } else {
            sskip = 0;
            float theta = 0.5f * (A[q*KM + q] - A[p*KM + p]) / apq;
            float tt = (theta >= 0.0f ? 1.0f : -1.0f) /
                       (fabsf(theta) + sqrtf(theta * theta + 1.0f));
            float c = 1.0f / sqrtf(tt * tt + 1.0f);
            sc = c; ss = tt * c;
          }
        }
        __syncthreads();
        if (!sskip) {                 // uniform branch (shared flag)
          float c = sc, s = ss;
          for (int j = t; j < KM; j += blockDim.x) {
            float ajp = A[j*KM + p], ajq = A[j*KM + q];
            A[j*KM + p] = c * ajp - s * ajq;
            A[j*KM + q] = s * ajp + c * ajq;
          }
          __syncthreads();
          for (int j = t; j < KM; j += blockDim.x) {
            float apj = A[p*KM + j], aqj = A[q*KM + j];
            A[p*KM + j] = c * apj - s * aqj;
            A[q*KM + j] = s * apj + c * aqj;
          }
          for (int j = t; j < KM; j += blockDim.x) {
            float qjp = Q[j*KM + p], qjq = Q[j*KM + q];
            Q[j*KM + p] = c * qjp - s * qjq;
            Q[j*KM + q] = s * qjp + c * qjq;
          }
        }
        __syncthreads();
      }
    }
  }
  for (int i = t; i < KM; i += blockDim.x) w[i] = A[i*KM + i];
}

// Sort eigenvalues ascending; Qs = Q with permuted columns. One block.
__global__ __launch_bounds__(128) void k_sortq(const float* __restrict__ w,
                                               const float* __restrict__ Q,
                                               float* __restrict__ Qs) {
  __shared__ int perm[KM];
  __shared__ float ws[KM];
  if (threadIdx.x == 0) {
    for (int i = 0; i < KM; ++i) { ws[i] = w[i]; perm[i] = i; }
    for (int i = 0; i < KM - 1; ++i) {
      int best = i;
      for (int j = i + 1; j < KM; ++j)
        if (ws[j] < ws[best]) best = j;
      float tv = ws[i]; ws[i] = ws[best]; ws[best] = tv;
      int tp = perm[i]; perm[i] = perm[best]; perm[best] = tp;
    }
  }
  __syncthreads();
  for (int t = threadIdx.x; t < KM * KM; t += blockDim.x) {
    int i = t / KM, j = t % KM;
    Qs[i*KM + j] = Q[i*KM + perm[j]];
  }
}

// Out(4096xKM) = Xin(4096xKM) @ Qs(KMxKM).  grid(32,7), 8 waves/block.
__global__ __launch_bounds__(256) void k_xq(const float* __restrict__ Xin,
                                            const float* __restrict__ Qs,
                                            float* __restrict__ Out) {
  __shared__ float tB[KM][16];
  const int lane = threadIdx.x & 31;
  const int wave = threadIdx.x >> 5;
  const int m    = lane & 15;
  const int half = lane >> 4;
  const int row0 = blockIdx.x * 128 + wave * 16;
  const int col0 = blockIdx.y * 16;
  for (int t = threadIdx.x; t < KM * 16; t += 256) {
    int kk = t >> 4, c = t & 15;
    tB[kk][c] = Qs[(size_t)kk * KM + col0 + c];
  }
  __syncthreads();
  v8f acc = {};
#pragma unroll
  for (int kk = 0; kk < KM; kk += 4) {
    v2f a = *reinterpret_cast<const v2f*>(Xin + (size_t)(row0 + m) * KM + kk + 2 * half);
    v2f b;
    b.x = tB[kk + 2 * half + 0][m];
    b.y = tB[kk + 2 * half + 1][m];
    acc = __builtin_amdgcn_wmma_f32_16x16x4_f32(false, a, false, b,
                                                (short)0, acc, false, false);
  }
#pragma unroll
  for (int v = 0; v < 8; ++v)
    Out[(size_t)(row0 + v + 8 * half) * KM + col0 + m] = acc[v];
}

// Column norms of first NMODES columns
__global__ __launch_bounds__(256) void k_colnorm(const float* __restrict__ Xf,
                                                 float* __restrict__ norms) {
  __shared__ float red[256];
  int c = blockIdx.x;
  float s = 0.0f;
  for (int r = threadIdx.x; r < NV; r += 256) {
    float v = Xf[(size_t)r * KM + c];
    s += v * v;
  }
  red[threadIdx.x] = s;
  __syncthreads();
  for (int o = 128; o > 0; o >>= 1) {
    if (threadIdx.x < o) red[threadIdx.x] += red[threadIdx.x + o];
    __syncthreads();
  }
  if (threadIdx.x == 0) norms[c] = fmaxf(sqrtf(red[0]), EPSF);
}

__global__ void k_writeout(const float* __restrict__ Xf,
                           const float* __restrict__ norms,
                           float* __restrict__ out) {
  int idx = blockIdx.x * blockDim.x + threadIdx.x;
  if (idx >= NV * NMODES) return;
  int r = idx / NMODES, c = idx % NMODES;
  out[idx] = Xf[(size_t)r * KM + c] / norms[c];
}

// ---------------------------------------------------------------------------
extern "C" void kernel_launch(void* const* d_in, const int* in_sizes, int n_in,
                              void* d_out, int out_size, void* d_ws, size_t ws_size,
                              hipStream_t stream) {
  (void)in_sizes; (void)n_in; (void)out_size; (void)ws_size;
  const float* V = (const float*)d_in[0];
  const int*   F = (const int*)d_in[1];

  float* ws    = (float*)d_ws;
  float* Lm    = ws;                                   // 4096*4096
  float* Xa    = Lm + (size_t)NV * NV;                 // 4096*112
  float* Xb    = Xa + (size_t)NV * KM;                 // 4096*112
  float* G     = Xb + (size_t)NV * KM;                 // 112*112
  float* Q     = G  + KM * KM;                         // 112*112
  float* Qs    = Q  + KM * KM;                         // 112*112
  float* w     = Qs + KM * KM;                         // 112
  float* norms = w  + KM;                              // 112
  float* shift = norms + KM;                           // 1

  // 1. init
  k_zero<<<2048, 256, 0, stream>>>(Lm, (long)NV * NV);
  k_zero<<<1, 32, 0, stream>>>(shift, 1);
  k_initX<<<(NV * KM + 255) / 256, 256, 0, stream>>>(Xa);

  // 2. assemble L, 3. shift bound
  k_scatter<<<NF / 256, 256, 0, stream>>>(V, F, Lm);
  k_rowshift<<<NV, 256, 0, stream>>>(Lm, shift);

  // initial orthonormalization of X
  k_atb<<<dim3(7, 7), 32, 0, stream>>>(Xa, Xa, G);
  k_chol<<<1, 128, 0, stream>>>(G);
  k_trisolve<<<NV / 128, 128, 0, stream>>>(Xa, G);

  // 4. subspace iteration on B = s*I - L
  float* Xc = Xa;
  float* Yc = Xb;
  for (int it = 0; it < ITERS; ++it) {
    k_gemmLX<<<dim3(32, 7), 256, 0, stream>>>(Lm, Xc, Yc, shift, -1.0f);
    k_atb<<<dim3(7, 7), 32, 0, stream>>>(Yc, Yc, G);
    k_chol<<<1, 128, 0, stream>>>(G);
    k_trisolve<<<NV / 128, 128, 0, stream>>>(Yc, G);
    float* tmp = Xc; Xc = Yc; Yc = tmp;
  }

  // 5. Rayleigh-Ritz: W = L*X, Ar = X^T W, eig, rotate, normalize
  k_gemmLX<<<dim3(32, 7), 256, 0, stream>>>(Lm, Xc, Yc, nullptr, 1.0f);
  k_atb<<<dim3(7, 7), 32, 0, stream>>>(Xc, Yc, G);
  k_jacobi<<<1, 128, 0, stream>>>(G, Q, w);
  k_sortq<<<1, 128, 0, stream>>>(w, Q, Qs);
  k_xq<<<dim3(32, 7), 256, 0, stream>>>(Xc, Qs, Yc);
  k_colnorm<<<NMODES, 256, 0, stream>>>(Yc, norms);
  k_writeout<<<(NV * NMODES + 255) / 256, 256, 0, stream>>>(Yc, norms, (float*)d_out);
}